// Echellogram_58325655880212
// MI455X (gfx1250) — compile-verified
//
#include <hip/hip_runtime.h>
#include <hip/hip_bf16.h>

typedef __attribute__((ext_vector_type(2))) float v2f;
typedef __attribute__((ext_vector_type(8))) float v8f;

#define NXC 1024
#define NYC 85
#define NAMPS 1500
#define NPIX (NXC * NYC)          // 87040
#define TILE 16
#define NTILES (NPIX / TILE)      // 5440
#define WAVES_PER_BLOCK 8
#define BLOCK_T (WAVES_PER_BLOCK * 32)
#define NBLOCKS (NTILES / WAVES_PER_BLOCK) // 680

#define WROW_STRIDE 1504          // padded row stride for weight rows
#define WZERO (2 * WROW_STRIDE)   // 3008: 8-float zero slot for inactive lanes

#define SIGMA_CUBE 0.42f
#define INV_SIG (1.0f / 0.42f)
#define INV_SQRT_2PI 0.3989422804014327f
#define PDF_CONST (INV_SQRT_2PI / SIGMA_CUBE)
#define SLIT 12.0f

__global__ __launch_bounds__(BLOCK_T) void echellogram_wmma_kernel(
    const int* __restrict__ g_index,
    const float* __restrict__ g_bkg,
    const float* __restrict__ g_scoef,       // 3
    const float* __restrict__ g_amps,        // 1500
    const float* __restrict__ g_smooth,
    const float* __restrict__ g_lamcoef,     // 4
    const float* __restrict__ g_pcoef,       // 2x2
    const float* __restrict__ g_srcamps,     // 1500
    const float* __restrict__ g_fid,         // 2
    const float* __restrict__ g_lamv,        // 1500
    float* __restrict__ out)
{
    __shared__ __align__(16) float s_w[WZERO + 8];   // amps | src | zero slot
    __shared__ __align__(16) float s_lamv[NAMPS + 4];

    // Stage operands in LDS, pre-scaled: weights * 1/(sigma*sqrt(2pi)),
    // lam_vector * 1/sigma (so the loop computes z = lam/sigma - lamv/sigma).
    for (int i = threadIdx.x; i < NAMPS; i += BLOCK_T) {
        s_w[i]               = g_amps[i]    * PDF_CONST;
        s_w[WROW_STRIDE + i] = g_srcamps[i] * PDF_CONST;
        s_lamv[i]            = g_lamv[i]    * INV_SIG;
    }
    if (threadIdx.x < 8) s_w[WZERO + threadIdx.x] = 0.0f;
    __syncthreads();

    const int wave = threadIdx.x >> 5;
    const int lane = threadIdx.x & 31;
    const int tile = blockIdx.x * WAVES_PER_BLOCK + wave;

    const int m     = lane & 15;          // A row == B column (pixel-in-tile)
    const int khalf = (lane >> 4) << 1;   // K sub-offset: 0 (lanes 0-15) / 2 (16-31)

    const int p = tile * TILE + m;        // flat pixel, out layout x*NY + y
    const int x = p / NYC;
    const int y = p - x * NYC;
    const float fx = (float)x;
    const float fy = (float)y;

    const float s0 = g_scoef[0], s1 = g_scoef[1], s2 = g_scoef[2];
    const float c0 = g_lamcoef[0], c1 = g_lamcoef[1], c2 = g_lamcoef[2], c3 = g_lamcoef[3];
    const float fid0 = g_fid[0], fid1 = g_fid[1];

    // Along-slit coordinate.
    const float ss = s1 * (fy - s0 - s2 * fx);

    // Wavelength surface lam(x,y), pre-scaled by 1/sigma.
    const float xn = (fx - 512.0f) * (1.0f / 512.0f);
    const float yn = (fy - 42.5f) * (1.0f / 42.5f);
    const float cx1 = fid1 * (1.0f + c1 * 0.01f) * 512.0f;
    const float cx2 = 1.0f + c2;
    const float lam = fid0 + c0 + cx1 * xn + cx2 * (2.0f * xn * xn - 1.0f) + c3 * yn;
    const float lamn = lam * INV_SIG;

    // Branchless A-operand addressing: active lanes (m<2) walk their weight
    // row; all other lanes park on the zero slot with stride 0.
    const float* wptr = &s_w[(m == 0) ? khalf
                           : (m == 1) ? (WROW_STRIDE + khalf)
                                      : WZERO];
    const int wstep = (m < 2) ? 8 : 0;    // floats advanced per k0 iteration
    const float* lptr = &s_lamv[khalf];

    v8f acc0 = {};
    v8f acc1 = {};

    // K loop: first 1496 of 1500 amps, 4 per WMMA, x2 unroll over two accumulators.
    for (int it = 0; it < NAMPS / 8; ++it) {
        const v2f lv0 = *(const v2f*)(lptr);
        const v2f lv1 = *(const v2f*)(lptr + 4);
        const v2f a0  = *(const v2f*)(wptr);
        const v2f a1  = *(const v2f*)(wptr + 4);   // zero-slot lanes read zeros

        v2f b0, b1;
        {
            const float z0 = lamn - lv0.x;
            const float z1 = lamn - lv0.y;
            b0.x = __expf(-0.5f * z0 * z0);
            b0.y = __expf(-0.5f * z1 * z1);
        }
        {
            const float z0 = lamn - lv1.x;
            const float z1 = lamn - lv1.y;
            b1.x = __expf(-0.5f * z0 * z0);
            b1.y = __expf(-0.5f * z1 * z1);
        }

        acc0 = __builtin_amdgcn_wmma_f32_16x16x4_f32(
            false, a0, false, b0, (short)0, acc0, false, false);
        acc1 = __builtin_amdgcn_wmma_f32_16x16x4_f32(
            false, a1, false, b1, (short)0, acc1, false, false);

        lptr += 8;
        wptr += wstep;
    }

    // Epilogue: D row M=0 (sky) is acc[0] of lanes 0-15, row M=1 (src) acc[1].
    if (lane < 16) {
        const float sky  = acc0[0] + acc1[0];   // pdf const already folded in
        const float srcm = acc0[1] + acc1[1];

        const float smoothness = g_smooth[0];
        const float inv_beta = __expf(-smoothness);
        const float t0 = ss * inv_beta;
        const float t1 = (SLIT - ss) * inv_beta;
        const float emask = (1.0f / (1.0f + __expf(-t0))) *
                            (1.0f / (1.0f + __expf(-t1)));

        const int idx = g_index[0];
        const float pc0 = g_pcoef[idx * 2 + 0];
        const float pc1 = g_pcoef[idx * 2 + 1];
        const float inv_sigma_p = __expf(-pc1);
        const float u = (ss - pc0) * inv_sigma_p;
        const float prof = INV_SQRT_2PI * inv_sigma_p * __expf(-0.5f * u * u);

        out[p] = emask * sky + prof * srcm + g_bkg[0];
    }
}

extern "C" void kernel_launch(void* const* d_in, const int* in_sizes, int n_in,
                              void* d_out, int out_size, void* d_ws, size_t ws_size,
                              hipStream_t stream) {
    (void)in_sizes; (void)n_in; (void)out_size; (void)d_ws; (void)ws_size;
    const int*   g_index  = (const int*)  d_in[0];
    const float* g_bkg    = (const float*)d_in[1];
    const float* g_scoef  = (const float*)d_in[2];
    const float* g_amps   = (const float*)d_in[3];
    const float* g_smooth = (const float*)d_in[4];
    const float* g_lamc   = (const float*)d_in[5];
    const float* g_pcoef  = (const float*)d_in[6];
    const float* g_srca   = (const float*)d_in[7];
    const float* g_fid    = (const float*)d_in[8];
    const float* g_lamv   = (const float*)d_in[9];
    float* out = (float*)d_out;

    hipLaunchKernelGGL(echellogram_wmma_kernel,
                       dim3(NBLOCKS), dim3(BLOCK_T), 0, stream,
                       g_index, g_bkg, g_scoef, g_amps, g_smooth,
                       g_lamc, g_pcoef, g_srca, g_fid, g_lamv, out);
}